// GraphRiskAnalyzer_30751965839774
// MI455X (gfx1250) — compile-verified
//
#include <hip/hip_runtime.h>
#include <math.h>

typedef __attribute__((ext_vector_type(2))) float v2f;
typedef __attribute__((ext_vector_type(8))) float v8f;

#define FLT_NEG_MAX -3.402823466e38f

__device__ __forceinline__ float lrelu02(float x){ return x > 0.f ? x : 0.2f*x; }

// monotonic float<->uint encoding for atomicMax-based segment max
__device__ __forceinline__ unsigned encf(float f){
  unsigned u = __float_as_uint(f);
  return (u & 0x80000000u) ? ~u : (u | 0x80000000u);
}
__device__ __forceinline__ float decf(unsigned u){
  return __uint_as_float((u & 0x80000000u) ? (u & 0x7FFFFFFFu) : ~u);
}

// ---------------- generic fills / elementwise ----------------
__global__ void k_fill(float* __restrict__ p, float v, long long cnt){
  long long i = (long long)blockIdx.x*blockDim.x + threadIdx.x;
  long long st = (long long)gridDim.x*blockDim.x;
  for (; i < cnt; i += st) p[i] = v;
}

__global__ void k_relu(float* __restrict__ p, long long cnt){
  long long i = (long long)blockIdx.x*blockDim.x + threadIdx.x;
  long long st = (long long)gridDim.x*blockDim.x;
  for (; i < cnt; i += st) p[i] = fmaxf(p[i], 0.f);
}

__global__ void k_bias_act(float* __restrict__ p, const float* __restrict__ b, int n, int act){
  long long cnt = (long long)n*128;
  long long i = (long long)blockIdx.x*blockDim.x + threadIdx.x;
  long long st = (long long)gridDim.x*blockDim.x;
  for (; i < cnt; i += st){
    float v = p[i] + b[(int)(i & 127)];
    p[i] = act ? fmaxf(v, 0.f) : v;
  }
}

// ---------------- degree / norm ----------------
__global__ void k_deg_scatter(const int* __restrict__ dst, float* __restrict__ deg, int e){
  long long i = (long long)blockIdx.x*blockDim.x + threadIdx.x;
  long long st = (long long)gridDim.x*blockDim.x;
  for (; i < e; i += st) atomicAdd(&deg[dst[i]], 1.f);
}
__global__ void k_rsqrt(float* __restrict__ p, int n){
  long long i = (long long)blockIdx.x*blockDim.x + threadIdx.x;
  long long st = (long long)gridDim.x*blockDim.x;
  for (; i < n; i += st) p[i] = rsqrtf(p[i]);
}

// ---------------- node projection (K=35) + LN + ReLU ----------------
__global__ void k_proj_ln(const float* __restrict__ x, const float* __restrict__ W,
                          const float* __restrict__ b, const float* __restrict__ g,
                          const float* __restrict__ beta, float* __restrict__ out, int n){
  __shared__ float xs[40];
  __shared__ float s1[128], s2[128];
  int node = blockIdx.x, tid = threadIdx.x;
  if (tid < 35) xs[tid] = x[(long long)node*35 + tid];
  __syncthreads();
  float acc = b[tid];
  for (int k = 0; k < 35; ++k) acc += xs[k]*W[k*128 + tid];
  s1[tid] = acc; s2[tid] = acc*acc; __syncthreads();
  for (int off = 64; off; off >>= 1){
    if (tid < off){ s1[tid] += s1[tid+off]; s2[tid] += s2[tid+off]; }
    __syncthreads();
  }
  float mean = s1[0]*(1.f/128.f);
  float var  = s2[0]*(1.f/128.f) - mean*mean;
  float y = g[tid]*(acc - mean)*rsqrtf(var + 1e-5f) + beta[tid];
  out[(long long)node*128 + tid] = fmaxf(y, 0.f);
}

// ---------------- WMMA fp32 GEMM: Y = act(concat(A0,A1) @ W + bias) ----------------
// A0:[R,K0], A1:[R,K1] (A1 may be null, K1=0), W:[K0+K1, Kout] row-major, Kout multiple of 16 (<=128)
// block: 256 threads (8 waves), 32 rows per block; wave w owns output cols [16w,16w+16)
// and computes TWO 16x16 row tiles per k-step, reusing each B fragment twice.
__global__ void k_gemm_wmma(const float* __restrict__ A0, const float* __restrict__ A1,
                            int K0, int K1,
                            const float* __restrict__ W, const float* __restrict__ bias,
                            float* __restrict__ Y, int R, int Kout, int act){
  const int K = K0 + K1;
  const int LDA = K + 1;                // padded stride (conflict-free LDS A reads)
  __shared__ float As[32*257];          // supports K <= 256 (32.9 KB of the WGP's 320KB LDS)
  int rowBase = blockIdx.x * 32;
  int tid = threadIdx.x;
  for (int i = tid; i < 32*K; i += 256){
    int r = i / K, c = i - r*K;
    int gr = rowBase + r;
    float v = 0.f;
    if (gr < R) v = (c < K0) ? A0[(long long)gr*K0 + c] : A1[(long long)gr*K1 + (c - K0)];
    As[r*LDA + c] = v;
  }
  __syncthreads();
  int wave = tid >> 5, lane = tid & 31;
  int cb = wave * 16;
  if (cb < Kout){
    int m   = lane & 15;                // row (A) / col (B,D) within tile
    int kk0 = (lane >> 4) << 1;         // lanes 0-15: K=0,1 ; lanes 16-31: K=2,3
    v8f acc0 = {0.f,0.f,0.f,0.f,0.f,0.f,0.f,0.f};
    v8f acc1 = {0.f,0.f,0.f,0.f,0.f,0.f,0.f,0.f};
    const float* as0 = &As[m*LDA];
    const float* as1 = &As[(16 + m)*LDA];
    for (int k = 0; k < K; k += 4){
      v2f bf;
      const float* wp = W + (long long)(k + kk0)*Kout + cb + m;
      bf.x = wp[0];
      bf.y = wp[Kout];
      v2f a0; a0.x = as0[k + kk0]; a0.y = as0[k + kk0 + 1];
      v2f a1; a1.x = as1[k + kk0]; a1.y = as1[k + kk0 + 1];
      acc0 = __builtin_amdgcn_wmma_f32_16x16x4_f32(false, a0, false, bf, (short)0, acc0, false, false);
      acc1 = __builtin_amdgcn_wmma_f32_16x16x4_f32(false, a1, false, bf, (short)0, acc1, false, false);
    }
    int col = cb + m;
    float bb = bias ? bias[col] : 0.f;
    int rb = (lane >> 4) * 8;           // D: vgpr j -> M=j (lanes 0-15) / M=j+8 (lanes 16-31)
    #pragma unroll
    for (int j = 0; j < 8; ++j){
      int row = rowBase + rb + j;
      if (row < R){
        float v = acc0[j] + bb;
        if (act) v = fmaxf(v, 0.f);
        Y[(long long)row*Kout + col] = v;
      }
      int row1 = row + 16;
      if (row1 < R){
        float v = acc1[j] + bb;
        if (act) v = fmaxf(v, 0.f);
        Y[(long long)row1*Kout + col] = v;
      }
    }
  }
}

// ---------------- GCN aggregation ----------------
__global__ void k_gcn_init(const float* __restrict__ h, const float* __restrict__ dinv,
                           const float* __restrict__ b, float* __restrict__ agg, int n){
  long long cnt = (long long)n*128;
  long long i = (long long)blockIdx.x*blockDim.x + threadIdx.x;
  long long st = (long long)gridDim.x*blockDim.x;
  for (; i < cnt; i += st){
    int node = (int)(i >> 7), c = (int)(i & 127);
    float dv = dinv[node];
    agg[i] = h[i]*dv*dv + b[c];         // self-loop edge + bias
  }
}
// 4 channels per thread: one index load + one float4 per 4 atomics
__global__ void k_gcn_scatter(const float* __restrict__ h, const float* __restrict__ dinv,
                              const int* __restrict__ src, const int* __restrict__ dst,
                              float* __restrict__ agg, int e){
  long long cnt = (long long)e*32;
  long long i = (long long)blockIdx.x*blockDim.x + threadIdx.x;
  long long st = (long long)gridDim.x*blockDim.x;
  for (; i < cnt; i += st){
    int ee = (int)(i >> 5), q = (int)(i & 31);
    int s = src[ee], d = dst[ee];
    float w = dinv[s]*dinv[d];
    const float4 hv = *(const float4*)(h + (long long)s*128 + q*4);
    float* ap = agg + (long long)d*128 + q*4;
    atomicAdd(ap + 0, hv.x*w);
    atomicAdd(ap + 1, hv.y*w);
    atomicAdd(ap + 2, hv.z*w);
    atomicAdd(ap + 3, hv.w*w);
  }
}

// ---------------- GAT ----------------
__global__ void k_gat_alpha(const float* __restrict__ h, const float* __restrict__ asrc,
                            const float* __restrict__ adst, float* __restrict__ als,
                            float* __restrict__ ald, int n, int heads, int ch){
  long long cnt = (long long)n*heads;
  long long i = (long long)blockIdx.x*blockDim.x + threadIdx.x;
  long long st = (long long)gridDim.x*blockDim.x;
  for (; i < cnt; i += st){
    int node = (int)(i/heads), hd = (int)(i - (long long)node*heads);
    const float* hp = h + (long long)node*128 + hd*ch;
    float s = 0.f, d = 0.f;
    for (int c = 0; c < ch; ++c){ float v = hp[c]; s += v*asrc[hd*ch + c]; d += v*adst[hd*ch + c]; }
    als[i] = s; ald[i] = d;
  }
}
__device__ __forceinline__ void edge_sd(const int* src, const int* dst, int e, int ee, int& s, int& d){
  if (ee < e){ s = src[ee]; d = dst[ee]; } else { s = d = ee - e; }   // appended self loops
}
__global__ void k_gat_max(const float* __restrict__ als, const float* __restrict__ ald,
                          const int* __restrict__ src, const int* __restrict__ dst,
                          unsigned* __restrict__ menc, int e, int n, int heads){
  long long cnt = (long long)(e + n)*heads;
  long long i = (long long)blockIdx.x*blockDim.x + threadIdx.x;
  long long st = (long long)gridDim.x*blockDim.x;
  for (; i < cnt; i += st){
    int ee = (int)(i/heads), hd = (int)(i - (long long)ee*heads);
    int s, d; edge_sd(src, dst, e, ee, s, d);
    float el = lrelu02(als[(long long)s*heads + hd] + ald[(long long)d*heads + hd]);
    atomicMax(&menc[(long long)d*heads + hd], encf(el));
  }
}
__global__ void k_gat_sum(const float* __restrict__ als, const float* __restrict__ ald,
                          const int* __restrict__ src, const int* __restrict__ dst,
                          const unsigned* __restrict__ menc, float* __restrict__ ssum,
                          int e, int n, int heads){
  long long cnt = (long long)(e + n)*heads;
  long long i = (long long)blockIdx.x*blockDim.x + threadIdx.x;
  long long st = (long long)gridDim.x*blockDim.x;
  for (; i < cnt; i += st){
    int ee = (int)(i/heads), hd = (int)(i - (long long)ee*heads);
    int s, d; edge_sd(src, dst, e, ee, s, d);
    long long ad = (long long)d*heads + hd;
    float el = lrelu02(als[(long long)s*heads + hd] + ald[ad]);
    atomicAdd(&ssum[ad], expf(el - decf(menc[ad])));
  }
}
// 4 channels per thread: alpha (incl. exp) computed once per 4 channels
__global__ void k_gat_out(const float* __restrict__ h, const float* __restrict__ als,
                          const float* __restrict__ ald, const unsigned* __restrict__ menc,
                          const float* __restrict__ ssum, const int* __restrict__ src,
                          const int* __restrict__ dst, float* __restrict__ out,
                          int e, int n, int heads, int ch){
  long long cnt = (long long)(e + n)*32;    // heads*ch == 128 always; 4 ch per thread
  long long i = (long long)blockIdx.x*blockDim.x + threadIdx.x;
  long long st = (long long)gridDim.x*blockDim.x;
  for (; i < cnt; i += st){
    int ee = (int)(i >> 5), q = (int)(i & 31);
    int hd = (q*4) / ch;                    // 4 | ch, so the 4 channels share one head
    int s, d; edge_sd(src, dst, e, ee, s, d);
    long long ad = (long long)d*heads + hd;
    float el = lrelu02(als[(long long)s*heads + hd] + ald[ad]);
    float alpha = expf(el - decf(menc[ad])) / (ssum[ad] + 1e-16f);
    const float4 hv = *(const float4*)(h + (long long)s*128 + q*4);
    float* op = out + (long long)d*128 + q*4;
    atomicAdd(op + 0, hv.x*alpha);
    atomicAdd(op + 1, hv.y*alpha);
    atomicAdd(op + 2, hv.z*alpha);
    atomicAdd(op + 3, hv.w*alpha);
  }
}

// ---------------- LN(+ReLU) over width 128 ----------------
__global__ void k_ln_relu(const float* __restrict__ in, const float* __restrict__ g,
                          const float* __restrict__ beta, float* __restrict__ out, int n){
  __shared__ float s1[128], s2[128];
  int node = blockIdx.x, tid = threadIdx.x;
  float v = in[(long long)node*128 + tid];
  s1[tid] = v; s2[tid] = v*v; __syncthreads();
  for (int off = 64; off; off >>= 1){
    if (tid < off){ s1[tid] += s1[tid+off]; s2[tid] += s2[tid+off]; }
    __syncthreads();
  }
  float mean = s1[0]*(1.f/128.f);
  float var  = s2[0]*(1.f/128.f) - mean*mean;
  float y = g[tid]*(v - mean)*rsqrtf(var + 1e-5f) + beta[tid];
  out[(long long)node*128 + tid] = fmaxf(y, 0.f);
}

// ---------------- risk head: [N,64] -> logits[N,3], probs[N,3] ----------------
__global__ void k_risk2(const float* __restrict__ r1, const float* __restrict__ W2,
                        const float* __restrict__ b2, float* __restrict__ logits,
                        float* __restrict__ probs, int n){
  long long i = (long long)blockIdx.x*blockDim.x + threadIdx.x;
  long long st = (long long)gridDim.x*blockDim.x;
  for (; i < n; i += st){
    const float* rp = r1 + i*64;
    float l0 = b2[0], l1 = b2[1], l2 = b2[2];
    for (int k = 0; k < 64; ++k){
      float v = rp[k];
      l0 += v*W2[k*3]; l1 += v*W2[k*3+1]; l2 += v*W2[k*3+2];
    }
    logits[i*3] = l0; logits[i*3+1] = l1; logits[i*3+2] = l2;
    float m = fmaxf(l0, fmaxf(l1, l2));
    float e0 = expf(l0-m), e1 = expf(l1-m), e2 = expf(l2-m);
    float s = e0 + e1 + e2;
    probs[i*3] = e0/s; probs[i*3+1] = e1/s; probs[i*3+2] = e2/s;
  }
}

// ---------------- graph pooling: mean/max per channel ----------------
__global__ void k_pool(const float* __restrict__ ne, float* __restrict__ gfeat, int n){
  __shared__ float ssum[256], smax[256];
  int c = blockIdx.x, tid = threadIdx.x;
  float s = 0.f, m = FLT_NEG_MAX;
  for (int r = tid; r < n; r += 256){
    float v = ne[(long long)r*128 + c];
    s += v; m = fmaxf(m, v);
  }
  ssum[tid] = s; smax[tid] = m; __syncthreads();
  for (int off = 128; off; off >>= 1){
    if (tid < off){ ssum[tid] += ssum[tid+off]; smax[tid] = fmaxf(smax[tid], smax[tid+off]); }
    __syncthreads();
  }
  if (tid == 0){ gfeat[c] = ssum[0]/(float)n; gfeat[128 + c] = smax[0]; }
}

// ---------------- graph head: [1,256]@[256,256] + LN + tanh ----------------
__global__ void k_graph_head(const float* __restrict__ gfeat, const float* __restrict__ W,
                             const float* __restrict__ b, const float* __restrict__ g,
                             const float* __restrict__ beta, float* __restrict__ out){
  __shared__ float gf[256], s1[256], s2[256];
  int tid = threadIdx.x;
  gf[tid] = gfeat[tid]; __syncthreads();
  float acc = b[tid];
  for (int k = 0; k < 256; ++k) acc += gf[k]*W[k*256 + tid];
  s1[tid] = acc; s2[tid] = acc*acc; __syncthreads();
  for (int off = 128; off; off >>= 1){
    if (tid < off){ s1[tid] += s1[tid+off]; s2[tid] += s2[tid+off]; }
    __syncthreads();
  }
  float mean = s1[0]*(1.f/256.f);
  float var  = s2[0]*(1.f/256.f) - mean*mean;
  out[tid] = tanhf(g[tid]*(acc - mean)*rsqrtf(var + 1e-5f) + beta[tid]);
}

// ---------------- edge predictor (fused gather + WMMA GEMM + MLP head) ----------------
// per block: 32 edges. ef = [ne[src]|ne[dst]] (32x256 in LDS), h1 = relu(ef@W1+b1) (32x128 in LDS),
// out = sigmoid(h1@W2+b2). Each wave reuses every B(W1) fragment for two row tiles.
__global__ void k_edge_pred(const float* __restrict__ ne, const int* __restrict__ src,
                            const int* __restrict__ dst, const float* __restrict__ W1,
                            const float* __restrict__ b1, const float* __restrict__ W2,
                            const float* __restrict__ b2, float* __restrict__ out, int e){
  __shared__ float ef[32*257];
  __shared__ float h1[32*129];
  int e0 = blockIdx.x*32, tid = threadIdx.x;
  for (int i = tid; i < 32*256; i += 256){
    int r = i >> 8, c = i & 255;
    int ee = e0 + r;
    float v = 0.f;
    if (ee < e){
      int node = (c < 128) ? src[ee] : dst[ee];
      v = ne[(long long)node*128 + (c & 127)];
    }
    ef[r*257 + c] = v;
  }
  __syncthreads();
  int lane = tid & 31, wave = tid >> 5;
  int cb = wave*16, m = lane & 15, kk0 = (lane >> 4) << 1;
  v8f acc0 = {0.f,0.f,0.f,0.f,0.f,0.f,0.f,0.f};
  v8f acc1 = {0.f,0.f,0.f,0.f,0.f,0.f,0.f,0.f};
  const float* as0 = &ef[m*257];
  const float* as1 = &ef[(16 + m)*257];
  for (int k = 0; k < 256; k += 4){
    v2f bf;
    const float* wp = W1 + (k + kk0)*128 + cb + m;
    bf.x = wp[0];
    bf.y = wp[128];
    v2f a0; a0.x = as0[k + kk0]; a0.y = as0[k + kk0 + 1];
    v2f a1; a1.x = as1[k + kk0]; a1.y = as1[k + kk0 + 1];
    acc0 = __builtin_amdgcn_wmma_f32_16x16x4_f32(false, a0, false, bf, (short)0, acc0, false, false);
    acc1 = __builtin_amdgcn_wmma_f32_16x16x4_f32(false, a1, false, bf, (short)0, acc1, false, false);
  }
  int col = cb + m, rb = (lane >> 4)*8;
  float bb = b1[col];
  #pragma unroll
  for (int j = 0; j < 8; ++j){
    h1[(rb + j)*129 + col]      = fmaxf(acc0[j] + bb, 0.f);
    h1[(16 + rb + j)*129 + col] = fmaxf(acc1[j] + bb, 0.f);
  }
  __syncthreads();
  if (tid < 32){
    int ee = e0 + tid;
    if (ee < e){
      float a2 = b2[0];
      const float* hp = &h1[tid*129];
      for (int k = 0; k < 128; ++k) a2 += hp[k]*W2[k];
      out[ee] = 1.f/(1.f + expf(-a2));
    }
  }
}

// ==================== host launch ====================
static inline unsigned gblk(long long cnt){
  long long b = (cnt + 255)/256;
  if (b > 1048576) b = 1048576;
  if (b < 1) b = 1;
  return (unsigned)b;
}

extern "C" void kernel_launch(void* const* d_in, const int* in_sizes, int n_in,
                              void* d_out, int out_size, void* d_ws, size_t ws_size,
                              hipStream_t stream){
  const float* x        = (const float*)d_in[0];
  const int*   ei       = (const int*)  d_in[1];
  const float* p_W      = (const float*)d_in[2];
  const float* p_b      = (const float*)d_in[3];
  const float* p_g      = (const float*)d_in[4];
  const float* p_beta   = (const float*)d_in[5];
  const float* gcn_W    = (const float*)d_in[6];
  const float* gcn_b    = (const float*)d_in[7];
  const float* gat_W    = (const float*)d_in[8];
  const float* gat_asrc = (const float*)d_in[9];
  const float* gat_adst = (const float*)d_in[10];
  const float* gat_b    = (const float*)d_in[11];
  const float* gat3_W   = (const float*)d_in[12];
  const float* gat3_as  = (const float*)d_in[13];
  const float* gat3_ad  = (const float*)d_in[14];
  const float* gat3_b   = (const float*)d_in[15];
  const float* fus_W    = (const float*)d_in[16];
  const float* fus_b    = (const float*)d_in[17];
  const float* fus_g    = (const float*)d_in[18];
  const float* fus_beta = (const float*)d_in[19];
  const float* risk_W1  = (const float*)d_in[20];
  const float* risk_b1  = (const float*)d_in[21];
  const float* risk_W2  = (const float*)d_in[22];
  const float* risk_b2  = (const float*)d_in[23];
  const float* gf_W     = (const float*)d_in[24];
  const float* gf_b     = (const float*)d_in[25];
  const float* gf_g     = (const float*)d_in[26];
  const float* gf_beta  = (const float*)d_in[27];
  const float* ep_W1    = (const float*)d_in[28];
  const float* ep_b1    = (const float*)d_in[29];
  const float* ep_W2    = (const float*)d_in[30];
  const float* ep_b2    = (const float*)d_in[31];

  const int n = in_sizes[0]/35;
  const int e = in_sizes[1]/2;
  const int* src = ei;
  const int* dst = ei + e;

  // workspace layout (floats) — all offsets stay 16B aligned
  float* ws = (float*)d_ws;
  long long o = 0;
  float*    dinv  = ws + o; o += n;
  float*    als   = ws + o; o += (long long)n*8;
  float*    ald   = ws + o; o += (long long)n*8;
  unsigned* menc  = (unsigned*)(ws + o); o += (long long)n*8;
  float*    ssum  = ws + o; o += (long long)n*8;
  float*    gfeat = ws + o; o += 256;
  float*    nf    = ws + o; o += (long long)n*128;   // also reused as D after GAT-L0
  float*    A     = ws + o; o += (long long)n*128;
  float*    B     = ws + o; o += (long long)n*128;
  float*    C     = ws + o; o += (long long)n*128;
  float*    D     = nf;

  float* out_ne     = (float*)d_out;
  float* out_gemb   = out_ne   + (long long)n*128;
  float* out_logits = out_gemb + 256;
  float* out_probs  = out_logits + (long long)n*3;
  float* out_edge   = out_probs  + (long long)n*3;

  const long long n128 = (long long)n*128;
  const int gemm_blocks = (n + 31)/32;

  // degree -> dinv (self loops counted via init to 1)
  k_fill<<<gblk(n), 256, 0, stream>>>(dinv, 1.f, n);
  k_deg_scatter<<<gblk(e), 256, 0, stream>>>(dst, dinv, e);
  k_rsqrt<<<gblk(n), 256, 0, stream>>>(dinv, n);

  // node projection + LN + ReLU -> nf
  k_proj_ln<<<n, 128, 0, stream>>>(x, p_W, p_b, p_g, p_beta, nf, n);

  // ---- GCN branch: L0: nf->A->B(relu); L1: B->A->C(relu); L2: C->A->B ----
  const float* gcn_in[3]  = { nf, B, C };
  float*       gcn_out[3] = { B, C, B };
  for (int L = 0; L < 3; ++L){
    k_gemm_wmma<<<gemm_blocks, 256, 0, stream>>>(gcn_in[L], nullptr, 128, 0,
        gcn_W + (long long)L*128*128, nullptr, A, n, 128, 0);
    k_gcn_init<<<gblk(n128), 256, 0, stream>>>(A, dinv, gcn_b + L*128, gcn_out[L], n);
    k_gcn_scatter<<<gblk((long long)e*32), 256, 0, stream>>>(A, dinv, src, dst, gcn_out[L], e);
    if (L < 2) k_relu<<<gblk(n128), 256, 0, stream>>>(gcn_out[L], n128);
  }
  // gcn_x == B

  // ---- GAT branch ----
  // L0 (heads=8,ch=16): nf -> A -> C (bias+relu)
  // L1 (heads=8,ch=16): C  -> A -> D (bias+relu)    [D aliases nf, dead by now]
  // L2 (heads=1,ch=128): D -> A -> C (bias)
  const float* gat_in[3]   = { nf, C, D };
  float*       gat_o[3]    = { C, D, C };
  const float* gW[3]  = { gat_W, gat_W + 128*128, gat3_W };
  const float* gAS[3] = { gat_asrc, gat_asrc + 128, gat3_as };
  const float* gAD[3] = { gat_adst, gat_adst + 128, gat3_ad };
  const float* gB[3]  = { gat_b, gat_b + 128, gat3_b };
  const int    gH[3]  = { 8, 8, 1 };
  const int    gC[3]  = { 16, 16, 128 };
  for (int L = 0; L < 3; ++L){
    int heads = gH[L], ch = gC[L];
    k_gemm_wmma<<<gemm_blocks, 256, 0, stream>>>(gat_in[L], nullptr, 128, 0,
        gW[L], nullptr, A, n, 128, 0);
    k_gat_alpha<<<gblk((long long)n*heads), 256, 0, stream>>>(A, gAS[L], gAD[L], als, ald, n, heads, ch);
    k_fill<<<gblk((long long)n*heads), 256, 0, stream>>>((float*)menc, 0.f, (long long)n*heads);
    k_fill<<<gblk((long long)n*heads), 256, 0, stream>>>(ssum, 0.f, (long long)n*heads);
    k_fill<<<gblk(n128), 256, 0, stream>>>(gat_o[L], 0.f, n128);
    k_gat_max<<<gblk((long long)(e+n)*heads), 256, 0, stream>>>(als, ald, src, dst, menc, e, n, heads);
    k_gat_sum<<<gblk((long long)(e+n)*heads), 256, 0, stream>>>(als, ald, src, dst, menc, ssum, e, n, heads);
    k_gat_out<<<gblk((long long)(e+n)*32), 256, 0, stream>>>(A, als, ald, menc, ssum, src, dst,
        gat_o[L], e, n, heads, ch);
    k_bias_act<<<gblk(n128), 256, 0, stream>>>(gat_o[L], gB[L], n, (L < 2) ? 1 : 0);
  }
  // gat_x == C

  // ---- fusion: concat(B,C) @ fus_W + b -> D, then LN+ReLU -> ne (d_out) ----
  k_gemm_wmma<<<gemm_blocks, 256, 0, stream>>>(B, C, 128, 128, fus_W, fus_b, D, n, 128, 0);
  k_ln_relu<<<n, 128, 0, stream>>>(D, fus_g, fus_beta, out_ne, n);

  // ---- risk head ----
  k_gemm_wmma<<<gemm_blocks, 256, 0, stream>>>(out_ne, nullptr, 128, 0, risk_W1, risk_b1, A, n, 64, 1);
  k_risk2<<<gblk(n), 256, 0, stream>>>(A, risk_W2, risk_b2, out_logits, out_probs, n);

  // ---- graph embedding ----
  k_pool<<<128, 256, 0, stream>>>(out_ne, gfeat, n);
  k_graph_head<<<1, 256, 0, stream>>>(gfeat, gf_W, gf_b, gf_g, gf_beta, out_gemb);

  // ---- edge strengths ----
  k_edge_pred<<<(e + 31)/32, 256, 0, stream>>>(out_ne, src, dst, ep_W1, ep_b1, ep_W2, ep_b2, out_edge, e);
}